// Decoder_83330955477077
// MI455X (gfx1250) — compile-verified
//
#include <hip/hip_runtime.h>
#include <cmath>

// MI455X / gfx1250 pointer-network decoder.
// Persistent per-block 63-step scan: 16 batch rows per workgroup (M=16 == WMMA
// tile height), fp32 WMMA (v_wmma_f32_16x16x4_f32) for all GEMMs, step-invariant
// attention/pointer projections hoisted into d_ws (L2-resident, 16 MB).

typedef __attribute__((ext_vector_type(2))) float v2f;
typedef __attribute__((ext_vector_type(8))) float v8f;

constexpr int B_   = 256;
constexpr int S_   = 64;
constexpr int H_   = 128;
constexpr int ZD_  = 128;
constexpr int IN_  = 2;
constexpr int BPB  = 16;   // batch rows per block  -> WMMA M = 16
constexpr int TPB  = 256;  // 8 waves of 32
constexpr int NWAV = TPB / 32;
constexpr int NSTEP = S_ - 1;

__device__ __forceinline__ v8f wmma4(v2f a, v2f b, v8f c) {
  // D = A(16x4,f32) * B(4x16,f32) + C(16x16,f32)
  return __builtin_amdgcn_wmma_f32_16x16x4_f32(
      /*neg_a=*/false, a, /*neg_b=*/false, b,
      /*c_mod=*/(short)0, c, /*reuse_a=*/false, /*reuse_b=*/false);
}

// Accumulate one 16x16 output tile: A is row-major [16 x K] (LDS or global),
// B comes from weight matrix W.
//   WT=true : B[k][n] = W[(ncol0+n)*ldw + k]   (i.e. X @ W^T, torch-style)
//   WT=false: B[k][n] = W[k*ldw + (ncol0+n)]   (i.e. X @ W)
template <bool WT>
__device__ __forceinline__ void mma_tile(v8f& acc, const float* __restrict__ A,
                                         int lda, const float* __restrict__ W,
                                         int ldw, int ncol0, int K, int lane) {
  const int Mrow = lane & 15;
  const int kk   = (lane >> 4) << 1;      // lanes 0-15: K pair {0,1}; 16-31: {2,3}
  const int N    = ncol0 + (lane & 15);
  const float* arow = A + Mrow * lda + kk;
  for (int k0 = 0; k0 < K; k0 += 4) {
    v2f a;
    a.x = arow[k0];
    a.y = arow[k0 + 1];
    v2f b;
    if (WT) {
      const float* wr = W + N * ldw + k0 + kk;
      b.x = wr[0];
      b.y = wr[1];
    } else {
      b.x = W[(k0 + kk) * ldw + N];
      b.y = W[(k0 + kk + 1) * ldw + N];
    }
    acc = wmma4(a, b, acc);
  }
}

// C/D layout: VGPR r -> lanes 0-15 hold (M=r, N=lane), lanes 16-31 (M=r+8, N=lane-16)
__device__ __forceinline__ void mma_store(const v8f& acc, float* __restrict__ out,
                                          int ldo, int col0, int lane,
                                          const float* __restrict__ bias1,
                                          const float* __restrict__ bias2) {
  const int N     = lane & 15;
  const int mbase = (lane >> 4) << 3;
  float badd = 0.f;
  if (bias1) badd += bias1[col0 + N];
  if (bias2) badd += bias2[col0 + N];
#pragma unroll
  for (int r = 0; r < 8; ++r) {
    out[(mbase + r) * ldo + col0 + N] = acc[r] + badd;
  }
}

__device__ __forceinline__ float sigmoidf_(float x) {
  return 1.f / (1.f + __expf(-x));
}

__global__ __launch_bounds__(TPB, 1) void pointer_decoder_kernel(
    const float* __restrict__ instance,  // (B,S,IN)
    const int*   __restrict__ sol,       // (B,S)
    const float* __restrict__ Zin,       // (B,ZD)
    const float* __restrict__ IH,        // (B,S,H)
    const float* __restrict__ last_hh,   // (B,H)
    const float* __restrict__ W_emb, const float* __restrict__ b_emb,
    const float* __restrict__ W_ih,  const float* __restrict__ W_hh,
    const float* __restrict__ b_ih,  const float* __restrict__ b_hh,
    const float* __restrict__ W_a,   const float* __restrict__ v_a,
    const float* __restrict__ W1,    const float* __restrict__ b1,
    const float* __restrict__ W2,    const float* __restrict__ b2,
    const float* __restrict__ Wp,    const float* __restrict__ vp,
    float* __restrict__ outf,            // [B*S idx | B*(S-1) logp]
    float* __restrict__ ws)              // Ea (B*S*H) | Ep (B*S*H)
{
  __shared__ float sh_h[BPB * H_];      //  8 KB  persistent GRU state
  __shared__ float sh_mask[BPB * S_];   //  4 KB  persistent mask
  __shared__ float sh_ref[BPB * IN_];   //         persistent ref_in
  __shared__ float bufA[BPB * 384];     // 24 KB  phase-reused arena
  __shared__ float bufB[BPB * 384];     // 24 KB  phase-reused arena

  const int tid   = threadIdx.x;
  const int lane  = tid & 31;
  const int wave  = tid >> 5;
  const int bbase = blockIdx.x * BPB;

  float* Ea = ws;                                   // IH @ W_a[:H]
  float* Ep = ws + (size_t)B_ * S_ * H_;            // IH @ Wp[:H]

  // ---------------- init persistent state ----------------
  for (int i = tid; i < BPB * H_; i += TPB)
    sh_h[i] = last_hh[(bbase + (i >> 7)) * H_ + (i & 127)];
  for (int i = tid; i < BPB * S_; i += TPB) {
    int bi = i >> 6, s = i & 63;
    sh_mask[i] = (s == sol[(bbase + bi) * S_]) ? 0.f : 1.f;
  }
  if (tid < BPB) {
    int b  = bbase + tid;
    int s0 = sol[b * S_];
    sh_ref[tid * 2 + 0] = instance[((size_t)b * S_ + s0) * IN_ + 0];
    sh_ref[tid * 2 + 1] = instance[((size_t)b * S_ + s0) * IN_ + 1];
    outf[b * S_ + 0] = (float)s0;                   // tour_idx[:,0] = solution[:,0]
  }

  // ------- hoisted step-invariant projections: Ea, Ep (once) -------
  {
    const float* Ablk = IH + (size_t)bbase * S_ * H_;   // (BPB*S) x H row-major
    const int TT = (BPB * S_ / 16) * (H_ / 16);         // 512 tiles per half
    for (int t = wave; t < 2 * TT; t += NWAV) {
      int half = (t >= TT) ? 1 : 0;
      int tt   = half ? t - TT : t;
      int tm   = tt >> 3, tn = tt & 7;
      const float* Wsel = half ? Wp : W_a;              // top H rows
      float* dst = (half ? Ep : Ea) + ((size_t)bbase * S_ + tm * 16) * H_;
      v8f acc = {};
      mma_tile<false>(acc, Ablk + tm * 16 * H_, H_, Wsel, H_, tn * 16, H_, lane);
      mma_store(acc, dst, H_, tn * 16, lane, nullptr, nullptr);
    }
  }
  __threadfence();
  __syncthreads();

  // ---------------- 63-step recurrence ----------------
  for (int j = 1; j <= NSTEP; ++j) {
    // (1) ref_h = ref_in @ W_emb + b_emb  -> bufB[0:2048]
    for (int i = tid; i < BPB * H_; i += TPB) {
      int m = i >> 7, c = i & 127;
      bufB[i] = sh_ref[m * 2] * W_emb[c] + sh_ref[m * 2 + 1] * W_emb[H_ + c] + b_emb[c];
    }
    __syncthreads();

    // (2) GRU gates via WMMA. gsum = ref_h@W_ih^T + h@W_hh^T + b_ih + b_hh
    //     (chained accumulator), hn = h@W_hh[2H:]^T + b_hh[2H:]
    for (int t = wave; t < 32; t += NWAV) {
      if (t < 24) {
        int col0 = t * 16;
        v8f acc = {};
        mma_tile<true>(acc, bufB, H_, W_ih, H_, col0, H_, lane);
        mma_tile<true>(acc, sh_h, H_, W_hh, H_, col0, H_, lane);
        mma_store(acc, bufA, 384, col0, lane, b_ih, b_hh);
      } else {
        int col0 = (t - 24) * 16;
        v8f acc = {};
        mma_tile<true>(acc, sh_h, H_, W_hh, H_, 256 + col0, H_, lane);
        mma_store(acc, bufB + 2048, H_, col0, lane, b_hh + 256, nullptr);
      }
    }
    __syncthreads();

    // (3) GRU elementwise: n = tanh(gsum_n + (r-1)*hn); h = (1-z)*n + z*h
    for (int i = tid; i < BPB * H_; i += TPB) {
      int m = i >> 7, c = i & 127;
      float r  = sigmoidf_(bufA[m * 384 + c]);
      float z  = sigmoidf_(bufA[m * 384 + 128 + c]);
      float gn = bufA[m * 384 + 256 + c];
      float hn = bufB[2048 + i];
      float n  = tanhf(gn + (r - 1.f) * hn);
      sh_h[i]  = (1.f - z) * n + z * sh_h[i];
    }
    __syncthreads();

    // (4) Ah = h @ W_a[H:]  -> bufA[0:2048]
    for (int t = wave; t < 8; t += NWAV) {
      int col0 = t * 16;
      v8f acc = {};
      mma_tile<false>(acc, sh_h, H_, W_a + H_ * H_, H_, col0, H_, lane);
      mma_store(acc, bufA, H_, col0, lane, nullptr, nullptr);
    }
    __syncthreads();

    // (5) attention scores e[bi][s] = sum_c v_a[c]*tanh(Ea + Ah) -> bufA[2048+row]
    for (int row = tid; row < BPB * S_; row += TPB) {
      int bi = row >> 6, s = row & 63;
      const float* ea = Ea + ((size_t)(bbase + bi) * S_ + s) * H_;
      __builtin_prefetch(Ep + ((size_t)(bbase + bi) * S_ + s) * H_, 0, 0);
      const float* ah = bufA + bi * H_;
      float acc = 0.f;
      for (int c = 0; c < H_; ++c) acc += v_a[c] * tanhf(ea[c] + ah[c]);
      bufA[2048 + row] = acc;
    }
    __syncthreads();

    // (6) softmax over S (serial per batch row; S=64)
    if (tid < BPB) {
      float* e = bufA + 2048 + tid * S_;
      float mx = -INFINITY;
      for (int s = 0; s < S_; ++s) mx = fmaxf(mx, e[s]);
      float sum = 0.f;
      for (int s = 0; s < S_; ++s) { float v = __expf(e[s] - mx); e[s] = v; sum += v; }
      float inv = 1.f / sum;
      for (int s = 0; s < S_; ++s) e[s] *= inv;
    }
    __syncthreads();

    // (7) fc_in = [context | Z | ref_h] -> bufB[16][384]
    for (int i = tid; i < BPB * 384; i += TPB) {
      int m = i / 384, c = i % 384;
      float v;
      if (c < 128) {
        const float* a  = bufA + 2048 + m * S_;
        const float* ih = IH + (size_t)(bbase + m) * S_ * H_ + c;
        float acc = 0.f;
        for (int s = 0; s < S_; ++s) acc += a[s] * ih[(size_t)s * H_];
        v = acc;
      } else if (c < 256) {
        v = Zin[(bbase + m) * ZD_ + (c - 128)];
      } else {
        int cc = c - 256;  // recompute ref_h (2 FMAs) instead of keeping it live
        v = sh_ref[m * 2] * W_emb[cc] + sh_ref[m * 2 + 1] * W_emb[H_ + cc] + b_emb[cc];
      }
      bufB[i] = v;
    }
    __syncthreads();

    // (8) h1 = fc_in @ W1 + b1 -> bufA[0:4096] (ldo=256)
    for (int t = wave; t < 16; t += NWAV) {
      int col0 = t * 16;
      v8f acc = {};
      mma_tile<false>(acc, bufB, 384, W1, 256, col0, 384, lane);
      mma_store(acc, bufA, 256, col0, lane, b1, nullptr);
    }
    __syncthreads();

    // (9) fco = h1 @ W2 + b2 -> bufA[4096:6144]
    for (int t = wave; t < 8; t += NWAV) {
      int col0 = t * 16;
      v8f acc = {};
      mma_tile<false>(acc, bufA, 256, W2, H_, col0, 256, lane);
      mma_store(acc, bufA + 4096, H_, col0, lane, b2, nullptr);
    }
    __syncthreads();

    // (10) Pf = fco @ Wp[H:] -> bufB[0:2048]
    for (int t = wave; t < 8; t += NWAV) {
      int col0 = t * 16;
      v8f acc = {};
      mma_tile<false>(acc, bufA + 4096, H_, Wp + H_ * H_, H_, col0, H_, lane);
      mma_store(acc, bufB, H_, col0, lane, nullptr, nullptr);
    }
    __syncthreads();

    // (11) masked logits x[bi][s] = sum_c vp[c]*tanh(Ep + Pf) + log(mask)
    for (int row = tid; row < BPB * S_; row += TPB) {
      int bi = row >> 6, s = row & 63;
      const float* ep = Ep + ((size_t)(bbase + bi) * S_ + s) * H_;
      const float* pf = bufB + bi * H_;
      float acc = 0.f;
      for (int c = 0; c < H_; ++c) acc += vp[c] * tanhf(ep[c] + pf[c]);
      bufB[2048 + row] = (sh_mask[row] > 0.f) ? acc : -INFINITY;
    }
    __syncthreads();

    // (12) log_softmax, emit pred/logp, update mask & ref_in
    if (tid < BPB) {
      int b = bbase + tid;
      float* x = bufB + 2048 + tid * S_;
      float mx = -INFINITY;
      int am = 0;
      for (int s = 0; s < S_; ++s) {
        if (x[s] > mx) { mx = x[s]; am = s; }        // first-max, as jnp.argmax
      }
      float sum = 0.f;
      for (int s = 0; s < S_; ++s) sum += __expf(x[s] - mx);
      float lse = __logf(sum);
      int ptr = sol[b * S_ + j];
      outf[B_ * S_ + b * NSTEP + (j - 1)] = x[ptr] - mx - lse;  // tour_logp
      outf[b * S_ + j] = (float)am;                             // tour_idx
      sh_mask[tid * S_ + ptr] = 0.f;                            // mask AFTER logprobs
      sh_ref[tid * 2 + 0] = instance[((size_t)b * S_ + ptr) * IN_ + 0];
      sh_ref[tid * 2 + 1] = instance[((size_t)b * S_ + ptr) * IN_ + 1];
    }
    __syncthreads();
  }
}

extern "C" void kernel_launch(void* const* d_in, const int* in_sizes, int n_in,
                              void* d_out, int out_size, void* d_ws, size_t ws_size,
                              hipStream_t stream) {
  const float* instance = (const float*)d_in[0];
  const int*   sol      = (const int*)  d_in[1];
  const float* Zin      = (const float*)d_in[2];
  const float* IH       = (const float*)d_in[3];
  const float* last_hh  = (const float*)d_in[4];
  const float* W_emb    = (const float*)d_in[5];
  const float* b_emb    = (const float*)d_in[6];
  const float* W_ih     = (const float*)d_in[7];
  const float* W_hh     = (const float*)d_in[8];
  const float* b_ih     = (const float*)d_in[9];
  const float* b_hh     = (const float*)d_in[10];
  const float* W_a      = (const float*)d_in[11];
  const float* v_a      = (const float*)d_in[12];
  const float* W1       = (const float*)d_in[13];
  const float* b1       = (const float*)d_in[14];
  const float* W2       = (const float*)d_in[15];
  const float* b2       = (const float*)d_in[16];
  const float* Wp       = (const float*)d_in[17];
  const float* vp       = (const float*)d_in[18];

  dim3 grid(B_ / BPB);   // 16 blocks, each owns 16 batch rows for all 63 steps
  dim3 block(TPB);
  pointer_decoder_kernel<<<grid, block, 0, stream>>>(
      instance, sol, Zin, IH, last_hh, W_emb, b_emb, W_ih, W_hh, b_ih, b_hh,
      W_a, v_a, W1, b1, W2, b2, Wp, vp, (float*)d_out, (float*)d_ws);
}